// ArchivalMemory_16801912062746
// MI455X (gfx1250) — compile-verified
//
#include <hip/hip_runtime.h>
#include <math.h>

// ---------------------------------------------------------------------------
// ArchivalMemory pipeline for MI455X (gfx1250, wave32)
// fp32 WMMA (V_WMMA_F32_16X16X4_F32) for all GEMM stages.
// TDM (tensor_load_to_lds) double-buffered staging of the key stream.
// ---------------------------------------------------------------------------

typedef float v2f __attribute__((ext_vector_type(2)));
typedef float v8f __attribute__((ext_vector_type(8)));
typedef unsigned int u32x4 __attribute__((ext_vector_type(4)));
typedef int i32x4 __attribute__((ext_vector_type(4)));
typedef int i32x8 __attribute__((ext_vector_type(8)));

#define B_SZ   512
#define N_KEYS 100000
#define D_SZ   1024
#define C_SZ   64
#define K_TOP  8
#define NSPLIT 8
#define CHUNK_TILES 8
#define CHUNK_KEYS  (CHUNK_TILES * 16)   // 128 keys per staged chunk

#if defined(__has_builtin)
#  if __has_builtin(__builtin_amdgcn_tensor_load_to_lds) && \
      __has_builtin(__builtin_amdgcn_s_wait_tensorcnt)
#    define HAVE_TDM 1
#  endif
#endif
#ifndef HAVE_TDM
#define HAVE_TDM 0
#endif

__device__ __forceinline__ v8f wmma_f32_k4(v2f a, v2f b, v8f c) {
  // D = A(16x4 f32) * B(4x16 f32) + C(16x16 f32)
  return __builtin_amdgcn_wmma_f32_16x16x4_f32(
      /*neg_a=*/false, a, /*neg_b=*/false, b,
      /*c_mod=*/(short)0, c, /*reuse_a=*/false, /*reuse_b=*/false);
}

__device__ __forceinline__ float gelu_erf(float x) {
  return 0.5f * x * (1.0f + erff(x * 0.70710678118654752f));
}

#if HAVE_TDM
// Issue a TDM load of `rows` consecutive 64-float key rows (row-major,
// stride 64) from gsrc into LDS byte offset lds_off. 2-D tile descriptor:
// tile = 64 x rows, tensor_dim0 = 64, tensor_dim0_stride = 64, data_size 4B.
__device__ __forceinline__ void tdm_load_key_rows(const float* gsrc,
                                                  unsigned lds_off, int rows) {
  unsigned long long ga = (unsigned long long)(uintptr_t)gsrc;
  u32x4 g0 = {};
  g0[0] = 1u;                                        // count = 1 (valid D#)
  g0[1] = lds_off;                                   // lds_addr (bytes)
  g0[2] = (unsigned)(ga & 0xffffffffu);              // global_addr[31:0]
  g0[3] = (unsigned)((ga >> 32) & 0x01ffffffu)       // global_addr[56:32]
        | (2u << 30);                                // type = 2 ("image")
  i32x8 g1 = {};
  g1[0] = (2 << 16);                   // workgroup_mask=0, data_size=2 (4B)
  g1[1] = (64 << 16);                  // tensor_dim0[15:0] = 64
  g1[2] = 0;                           // tensor_dim0 hi, tensor_dim1 lo = 0
  g1[3] = 0x0010 | (64 << 16);         // tensor_dim1 = 0x100000, tile_dim0=64
  g1[4] = rows & 0xffff;               // tile_dim1 = rows, tile_dim2 = 0
  g1[5] = 64;                          // tensor_dim0_stride = 64
  g1[6] = 0;
  g1[7] = 0;
  i32x4 gz = {};
#if __has_include(<hip/amd_detail/amd_gfx1250_TDM.h>)
  i32x8 gz8 = {};
  __builtin_amdgcn_tensor_load_to_lds(g0, g1, gz, gz, gz8, 0);
#else
  __builtin_amdgcn_tensor_load_to_lds(g0, g1, gz, gz, 0);
#endif
}
#endif  // HAVE_TDM

// ---------------------------------------------------------------------------
// Kernel 1: encoder  q = L3(L2(L1(query)))   [16 rows per block, 4 waves]
// ---------------------------------------------------------------------------
__global__ __launch_bounds__(128)
void encoder_kernel(const float* __restrict__ query,
                    const float* __restrict__ ew1, const float* __restrict__ eb1,
                    const float* __restrict__ ew2, const float* __restrict__ eb2,
                    const float* __restrict__ ew3, const float* __restrict__ eb3,
                    float* __restrict__ qc, float* __restrict__ qsq) {
  __shared__ float sH1[16 * 256];
  __shared__ float sH2[16 * 128];
  __shared__ float sQ3[16 * 64];

  const int tid   = threadIdx.x;
  const int wave  = tid >> 5;
  const int lane  = tid & 31;
  const int lrow  = lane & 15;
  const int hi    = lane >> 4;
  const int khalf = hi * 2;
  const int rowbase = blockIdx.x * 16;

  // ---- layer 1: K=1024, Ncols=256 ----
  for (int nt = wave; nt < 16; nt += 4) {
    const int ncol = nt * 16 + lrow;
    v8f c = {};
    for (int k = 0; k < 1024; k += 4) {
      const int kk = k + khalf;
      v2f a = *(const v2f*)(query + (size_t)(rowbase + lrow) * 1024 + kk);
      v2f b;
      b.x = ew1[(size_t)kk * 256 + ncol];
      b.y = ew1[(size_t)(kk + 1) * 256 + ncol];
      c = wmma_f32_k4(a, b, c);
    }
    const float bias = eb1[ncol];
#pragma unroll
    for (int v = 0; v < 8; ++v) {
      const int m = v + hi * 8;
      sH1[m * 256 + ncol] = gelu_erf(c[v] + bias);
    }
  }
  __syncthreads();

  // ---- layer 2: K=256, Ncols=128 ----
  for (int nt = wave; nt < 8; nt += 4) {
    const int ncol = nt * 16 + lrow;
    v8f c = {};
    for (int k = 0; k < 256; k += 4) {
      const int kk = k + khalf;
      v2f a = *(const v2f*)(sH1 + lrow * 256 + kk);
      v2f b;
      b.x = ew2[(size_t)kk * 128 + ncol];
      b.y = ew2[(size_t)(kk + 1) * 128 + ncol];
      c = wmma_f32_k4(a, b, c);
    }
    const float bias = eb2[ncol];
#pragma unroll
    for (int v = 0; v < 8; ++v) {
      const int m = v + hi * 8;
      sH2[m * 128 + ncol] = gelu_erf(c[v] + bias);
    }
  }
  __syncthreads();

  // ---- layer 3: K=128, Ncols=64 ----
  for (int nt = wave; nt < 4; nt += 4) {
    const int ncol = nt * 16 + lrow;
    v8f c = {};
    for (int k = 0; k < 128; k += 4) {
      const int kk = k + khalf;
      v2f a = *(const v2f*)(sH2 + lrow * 128 + kk);
      v2f b;
      b.x = ew3[(size_t)kk * 64 + ncol];
      b.y = ew3[(size_t)(kk + 1) * 64 + ncol];
      c = wmma_f32_k4(a, b, c);
    }
    const float bias = eb3[ncol];
#pragma unroll
    for (int v = 0; v < 8; ++v) {
      const int m = v + hi * 8;
      const float val = c[v] + bias;
      sQ3[m * 64 + ncol] = val;
      qc[(size_t)(rowbase + m) * 64 + ncol] = val;
    }
  }
  __syncthreads();

  if (tid < 16) {
    float s = 0.0f;
#pragma unroll 16
    for (int cc = 0; cc < 64; ++cc) {
      const float v = sQ3[tid * 64 + cc];
      s += v * v;
    }
    qsq[rowbase + tid] = s;
  }
}

// ---------------------------------------------------------------------------
// Kernel 2: per-key squared norms
// ---------------------------------------------------------------------------
__global__ __launch_bounds__(256)
void keysq_kernel(const float* __restrict__ keys, float* __restrict__ ksq) {
  const int n = blockIdx.x * 256 + threadIdx.x;
  if (n < N_KEYS) {
    const float4* kp = (const float4*)(keys + (size_t)n * 64);
    float s = 0.0f;
#pragma unroll
    for (int i = 0; i < 16; ++i) {
      const float4 v = kp[i];
      s += v.x * v.x + v.y * v.y + v.z * v.z + v.w * v.w;
    }
    ksq[n] = s;
  }
}

// ---------------------------------------------------------------------------
// Kernel 3: distance GEMM + per-split top-8.
// grid = (32 row-tiles, 8 key-splits), 8 waves. TDM double-buffered staging
// of key chunks into LDS; WMMA from LDS; per-lane sorted top-8 insertion.
// ---------------------------------------------------------------------------
#define TK_LDS_FLOATS (2 * CHUNK_KEYS * 64 + 8 * 16 * 17 + 16 * 128 + 16 * 128)

__global__ __launch_bounds__(256)
void topk_kernel(const float* __restrict__ qc, const float* __restrict__ qsq,
                 const float* __restrict__ keys, const float* __restrict__ ksq,
                 float* __restrict__ cand_d, int* __restrict__ cand_i) {
  extern __shared__ float smem[];
  float* sKeys  = smem;                              // 2 x 128 x 64
  float* sTile  = smem + 2 * CHUNK_KEYS * 64;        // 8 x 16 x 17
  float* sCandD = sTile + 8 * 16 * 17;               // 16 x 128
  int*   sCandI = (int*)(sCandD + 16 * 128);         // 16 x 128

  const int tid   = threadIdx.x;
  const int wave  = tid >> 5;
  const int lane  = tid & 31;
  const int lrow  = lane & 15;
  const int hi    = lane >> 4;
  const int khalf = hi * 2;
  const int rowbase = blockIdx.x * 16;
  const int split   = blockIdx.y;

  const int ntiles_tot = N_KEYS / 16;                         // 6250
  const int t0 = (ntiles_tot * split) / NSPLIT;
  const int t1 = (ntiles_tot * (split + 1)) / NSPLIT;

  // Preload A fragments of the 16x64 q tile (reused for every key tile)
  v2f areg[16];
#pragma unroll
  for (int kt = 0; kt < 16; ++kt) {
    areg[kt] = *(const v2f*)(qc + (size_t)(rowbase + lrow) * 64 + kt * 4 + khalf);
  }
  const float myqsq = qsq[rowbase + lrow];

  float bestd[K_TOP];
  int   besti[K_TOP];
#pragma unroll
  for (int e = 0; e < K_TOP; ++e) { bestd[e] = 3.4e38f; besti[e] = 0; }

  float* myTile = &sTile[wave * 16 * 17];

#if HAVE_TDM
  const int nchunks = (t1 - t0 + CHUNK_TILES - 1) / CHUNK_TILES;
  // prologue: issue chunk 0 into buffer 0
  if (wave == 0) {
    const int rows0 = min(CHUNK_TILES, t1 - t0) * 16;
    tdm_load_key_rows(keys + (size_t)t0 * 16 * 64,
                      (unsigned)(uintptr_t)sKeys, rows0);
  }
  for (int cchunk = 0; cchunk < nchunks; ++cchunk) {
    const int ctile = t0 + cchunk * CHUNK_TILES;
    const int tiles_here = min(CHUNK_TILES, t1 - ctile);
    __syncthreads();   // next-buffer consumers from chunk-2 are done
    if (wave == 0) {
      if (cchunk + 1 < nchunks) {
        const int ntile2 = t0 + (cchunk + 1) * CHUNK_TILES;
        const int rows2 = min(CHUNK_TILES, t1 - ntile2) * 16;
        float* nbuf = sKeys + ((cchunk + 1) & 1) * CHUNK_KEYS * 64;
        tdm_load_key_rows(keys + (size_t)ntile2 * 16 * 64,
                          (unsigned)(uintptr_t)nbuf, rows2);
        __builtin_amdgcn_s_wait_tensorcnt(1);   // chunk c complete (in-order)
      } else {
        __builtin_amdgcn_s_wait_tensorcnt(0);
      }
    }
    __syncthreads();   // staged chunk visible to all waves
    if (wave < tiles_here) {
      const float* kbuf = sKeys + (cchunk & 1) * CHUNK_KEYS * 64;
      const int nbase = (ctile + wave) * 16;
      v8f c = {};
#pragma unroll
      for (int kt = 0; kt < 16; ++kt) {
        v2f b = *(const v2f*)(kbuf + (wave * 16 + lrow) * 64 + kt * 4 + khalf);
        c = wmma_f32_k4(areg[kt], b, c);
      }
      const float kq = ksq[nbase + lrow];
#pragma unroll
      for (int v = 0; v < 8; ++v) {
        const int m = v + hi * 8;
        myTile[m * 17 + lrow] = kq - 2.0f * c[v];
      }
      __builtin_amdgcn_wave_barrier();
#pragma unroll
      for (int e = 0; e < 8; ++e) {
        const int ncol = hi * 8 + e;
        float d2 = myqsq + myTile[lrow * 17 + ncol];
        d2 = fmaxf(d2, 0.0f);
        const int gidx = nbase + ncol;
        if (d2 < bestd[K_TOP - 1]) {
          bestd[K_TOP - 1] = d2;
          besti[K_TOP - 1] = gidx;
#pragma unroll
          for (int j = K_TOP - 1; j > 0; --j) {
            if (bestd[j] < bestd[j - 1]) {
              const float td = bestd[j]; bestd[j] = bestd[j - 1]; bestd[j - 1] = td;
              const int   ti = besti[j]; besti[j] = besti[j - 1]; besti[j - 1] = ti;
            }
          }
        }
      }
      __builtin_amdgcn_wave_barrier();
    }
  }
#else
  // Fallback: direct global loads of key fragments (round-1 path)
  for (int nt = t0 + wave; nt < t1; nt += 8) {
    const int nbase = nt * 16;
    const int pn = nbase + 8 * 16;
    if (pn + 16 <= N_KEYS) {
      __builtin_prefetch(keys + (size_t)(pn + lrow) * 64, 0, 1);
    }
    v8f c = {};
#pragma unroll
    for (int kt = 0; kt < 16; ++kt) {
      v2f b = *(const v2f*)(keys + (size_t)(nbase + lrow) * 64 + kt * 4 + khalf);
      c = wmma_f32_k4(areg[kt], b, c);
    }
    const float kq = ksq[nbase + lrow];
#pragma unroll
    for (int v = 0; v < 8; ++v) {
      const int m = v + hi * 8;
      myTile[m * 17 + lrow] = kq - 2.0f * c[v];
    }
    __builtin_amdgcn_wave_barrier();
#pragma unroll
    for (int e = 0; e < 8; ++e) {
      const int ncol = hi * 8 + e;
      float d2 = myqsq + myTile[lrow * 17 + ncol];
      d2 = fmaxf(d2, 0.0f);
      const int gidx = nbase + ncol;
      if (d2 < bestd[K_TOP - 1]) {
        bestd[K_TOP - 1] = d2;
        besti[K_TOP - 1] = gidx;
#pragma unroll
        for (int j = K_TOP - 1; j > 0; --j) {
          if (bestd[j] < bestd[j - 1]) {
            const float td = bestd[j]; bestd[j] = bestd[j - 1]; bestd[j - 1] = td;
            const int   ti = besti[j]; besti[j] = besti[j - 1]; besti[j - 1] = ti;
          }
        }
      }
    }
    __builtin_amdgcn_wave_barrier();
  }
#endif

  // merge 16 partial sorted lists per row -> this split's top-8
  const int slot = wave * 2 + hi;   // 0..15
#pragma unroll
  for (int e = 0; e < K_TOP; ++e) {
    sCandD[lrow * 128 + slot * 8 + e] = bestd[e];
    sCandI[lrow * 128 + slot * 8 + e] = besti[e];
  }
  __syncthreads();

  if (tid < 16) {
    const int row = rowbase + tid;
    float dsel[K_TOP];
    int   isel[K_TOP];
#pragma unroll
    for (int e = 0; e < K_TOP; ++e) {
      float mn = 3.4e38f;
      int mi = 0;
      for (int cc = 0; cc < 128; ++cc) {
        const float v = sCandD[tid * 128 + cc];
        if (v < mn) { mn = v; mi = cc; }
      }
      dsel[e] = mn;
      isel[e] = sCandI[tid * 128 + mi];
      sCandD[tid * 128 + mi] = 3.4e38f;
    }
#pragma unroll
    for (int e = 0; e < K_TOP; ++e) {
      cand_d[((size_t)row * NSPLIT + split) * K_TOP + e] = dsel[e];
      cand_i[((size_t)row * NSPLIT + split) * K_TOP + e] = isel[e];
    }
  }
}

// ---------------------------------------------------------------------------
// Kernel 3b: merge per-split candidates -> final sorted top-8 + confidence
// ---------------------------------------------------------------------------
__global__ __launch_bounds__(256)
void merge_kernel(const float* __restrict__ cand_d, const int* __restrict__ cand_i,
                  float* __restrict__ topk_dist, int* __restrict__ topk_idx,
                  float* __restrict__ conf_out) {
  const int row = blockIdx.x * 256 + threadIdx.x;
  if (row >= B_SZ) return;
  float bestd[K_TOP];
  int   besti[K_TOP];
#pragma unroll
  for (int e = 0; e < K_TOP; ++e) { bestd[e] = 3.4e38f; besti[e] = 0; }
  for (int cc = 0; cc < NSPLIT * K_TOP; ++cc) {
    const float d = cand_d[(size_t)row * NSPLIT * K_TOP + cc];
    const int   i = cand_i[(size_t)row * NSPLIT * K_TOP + cc];
    if (d < bestd[K_TOP - 1]) {
      bestd[K_TOP - 1] = d;
      besti[K_TOP - 1] = i;
#pragma unroll
      for (int j = K_TOP - 1; j > 0; --j) {
        if (bestd[j] < bestd[j - 1]) {
          const float td = bestd[j]; bestd[j] = bestd[j - 1]; bestd[j - 1] = td;
          const int   ti = besti[j]; besti[j] = besti[j - 1]; besti[j - 1] = ti;
        }
      }
    }
  }
#pragma unroll
  for (int e = 0; e < K_TOP; ++e) {
    topk_dist[row * K_TOP + e] = bestd[e];
    topk_idx[row * K_TOP + e]  = besti[e];
  }
  conf_out[row] = 1.0f / (bestd[0] + 1e-6f);
}

// ---------------------------------------------------------------------------
// Kernel 4: decoder + inverse-distance weighted combine (unchanged).
// ---------------------------------------------------------------------------
#define DEC_LDS_FLOATS (16 * 1088 + 16 * 256 + 16 * 512 + 32)

__global__ __launch_bounds__(256)
void decoder_kernel(const float* __restrict__ keys, const float* __restrict__ context,
                    const float* __restrict__ dw1, const float* __restrict__ db1,
                    const float* __restrict__ dw2, const float* __restrict__ db2,
                    const float* __restrict__ dw3, const float* __restrict__ db3,
                    const float* __restrict__ topk_dist, const int* __restrict__ topk_idx,
                    float* __restrict__ out_combined) {
  extern __shared__ float smem[];
  float* sIn = smem;                     // 16 x 1088
  float* sH1 = smem + 16 * 1088;         // 16 x 256
  float* sH2 = sH1 + 16 * 256;           // 16 x 512
  float* sWn = sH2 + 16 * 512;           // 16 normalized weights
  float* sWt = sWn + 16;                 // 16 raw 1/(d+eps)

  const int tid   = threadIdx.x;
  const int wave  = tid >> 5;
  const int lane  = tid & 31;
  const int lrow  = lane & 15;
  const int hi    = lane >> 4;
  const int khalf = hi * 2;
  const int rowbase = blockIdx.x * 16;   // (b,j) rows, b = row>>3

  for (int i = tid; i < 16 * 1088; i += 256) {
    const int r = i / 1088;
    const int ccol = i - r * 1088;
    const int grow = rowbase + r;
    float v;
    if (ccol < 64) {
      const int key_id = topk_idx[grow];
      v = keys[(size_t)key_id * 64 + ccol];
    } else {
      v = context[(size_t)(grow >> 3) * 1024 + (ccol - 64)];
    }
    sIn[r * 1088 + ccol] = v;
  }
  if (tid < 16) {
    sWt[tid] = 1.0f / (topk_dist[rowbase + tid] + 1e-6f);
  }
  __syncthreads();
  if (tid < 16) {
    const int g = tid >> 3;
    float s = 0.0f;
#pragma unroll
    for (int j = 0; j < 8; ++j) s += sWt[g * 8 + j];
    sWn[tid] = sWt[tid] / s;
  }

  // ---- layer 1: K=1088, Ncols=256 ----
  for (int nt = wave; nt < 16; nt += 8) {
    const int ncol = nt * 16 + lrow;
    v8f c = {};
    for (int k = 0; k < 1088; k += 4) {
      const int kk = k + khalf;
      v2f a = *(const v2f*)(sIn + lrow * 1088 + kk);
      v2f b;
      b.x = dw1[(size_t)kk * 256 + ncol];
      b.y = dw1[(size_t)(kk + 1) * 256 + ncol];
      c = wmma_f32_k4(a, b, c);
    }
    const float bias = db1[ncol];
#pragma unroll
    for (int v = 0; v < 8; ++v) {
      const int m = v + hi * 8;
      sH1[m * 256 + ncol] = gelu_erf(c[v] + bias);
    }
  }
  __syncthreads();

  // ---- layer 2: K=256, Ncols=512 ----
  for (int nt = wave; nt < 32; nt += 8) {
    const int ncol = nt * 16 + lrow;
    v8f c = {};
    for (int k = 0; k < 256; k += 4) {
      const int kk = k + khalf;
      v2f a = *(const v2f*)(sH1 + lrow * 256 + kk);
      v2f b;
      b.x = dw2[(size_t)kk * 512 + ncol];
      b.y = dw2[(size_t)(kk + 1) * 512 + ncol];
      c = wmma_f32_k4(a, b, c);
    }
    const float bias = db2[ncol];
#pragma unroll
    for (int v = 0; v < 8; ++v) {
      const int m = v + hi * 8;
      sH2[m * 512 + ncol] = gelu_erf(c[v] + bias);
    }
  }
  __syncthreads();

  // ---- layer 3 + weighted combine: K=512, Ncols=1024 ----
  const int b0 = rowbase >> 3;
  for (int nt = wave; nt < 64; nt += 8) {
    const int ncol = nt * 16 + lrow;
    v8f c = {};
    for (int k = 0; k < 512; k += 4) {
      const int kk = k + khalf;
      v2f a = *(const v2f*)(sH2 + lrow * 512 + kk);
      v2f b;
      b.x = dw3[(size_t)kk * 1024 + ncol];
      b.y = dw3[(size_t)(kk + 1) * 1024 + ncol];
      c = wmma_f32_k4(a, b, c);
    }
    const float bias = db3[ncol];
    float acc = 0.0f;
#pragma unroll
    for (int v = 0; v < 8; ++v) {
      const int m = v + hi * 8;
      acc += (c[v] + bias) * sWn[m];
    }
    out_combined[(size_t)(b0 + hi) * 1024 + ncol] = acc;
  }
}

// ---------------------------------------------------------------------------
// Host launcher
// inputs: 0 query, 1 context, 2 keys, 3 ew1, 4 eb1, 5 ew2, 6 eb2, 7 ew3,
//         8 eb3, 9 dw1, 10 db1, 11 dw2, 12 db2, 13 dw3, 14 db3, 15 k(=8)
// d_out: combined [512*1024] then confidence [512]
// ---------------------------------------------------------------------------
extern "C" void kernel_launch(void* const* d_in, const int* in_sizes, int n_in,
                              void* d_out, int out_size, void* d_ws, size_t ws_size,
                              hipStream_t stream) {
  const float* query   = (const float*)d_in[0];
  const float* context = (const float*)d_in[1];
  const float* keys    = (const float*)d_in[2];
  const float* ew1 = (const float*)d_in[3];
  const float* eb1 = (const float*)d_in[4];
  const float* ew2 = (const float*)d_in[5];
  const float* eb2 = (const float*)d_in[6];
  const float* ew3 = (const float*)d_in[7];
  const float* eb3 = (const float*)d_in[8];
  const float* dw1 = (const float*)d_in[9];
  const float* db1 = (const float*)d_in[10];
  const float* dw2 = (const float*)d_in[11];
  const float* db2 = (const float*)d_in[12];
  const float* dw3 = (const float*)d_in[13];
  const float* db3 = (const float*)d_in[14];

  // workspace carve-up (floats)
  float* ws     = (float*)d_ws;
  float* qc     = ws;                               // 512*64
  float* qsq    = qc + B_SZ * C_SZ;                 // 512
  float* ksq    = qsq + B_SZ;                       // 100000
  float* tdist  = ksq + N_KEYS;                     // 512*8
  int*   tidx   = (int*)(tdist + B_SZ * K_TOP);     // 512*8
  float* cand_d = (float*)(tidx + B_SZ * K_TOP);    // 512*8*8
  int*   cand_i = (int*)(cand_d + B_SZ * NSPLIT * K_TOP); // 512*8*8

  float* out_combined = (float*)d_out;              // [512,1024]
  float* out_conf     = out_combined + B_SZ * D_SZ; // [512]

  encoder_kernel<<<B_SZ / 16, 128, 0, stream>>>(query, ew1, eb1, ew2, eb2,
                                                ew3, eb3, qc, qsq);
  keysq_kernel<<<(N_KEYS + 255) / 256, 256, 0, stream>>>(keys, ksq);

  dim3 tkgrid(B_SZ / 16, NSPLIT);
  topk_kernel<<<tkgrid, 256, TK_LDS_FLOATS * sizeof(float), stream>>>(
      qc, qsq, keys, ksq, cand_d, cand_i);
  merge_kernel<<<(B_SZ + 255) / 256, 256, 0, stream>>>(cand_d, cand_i,
                                                       tdist, tidx, out_conf);
  decoder_kernel<<<(B_SZ * K_TOP) / 16, 256, DEC_LDS_FLOATS * sizeof(float),
                   stream>>>(keys, context, dw1, db1, dw2, db2, dw3, db3,
                             tdist, tidx, out_combined);
}